// SmallRNNModel_21629455303255
// MI455X (gfx1250) — compile-verified
//
#include <hip/hip_runtime.h>
#include <math.h>

// Problem constants (match the JAX reference)
#define B_ 64
#define T_ 512
#define I_ 256
#define H_ 512
#define O_ 256

typedef __attribute__((ext_vector_type(2))) float v2f;
typedef __attribute__((ext_vector_type(8))) float v8f;
typedef __attribute__((ext_vector_type(4))) unsigned int u32x4;
typedef __attribute__((ext_vector_type(8))) int i32x8;
typedef __attribute__((ext_vector_type(4))) int i32x4;

// --------------------------------------------------------------------------
// CDNA5 f32 WMMA: D(16x16) = A(16x4) * B(4x16) + C, full f32 precision.
// --------------------------------------------------------------------------
__device__ __forceinline__ v8f wmma_f32_k4(v2f a, v2f b, v8f c) {
  return __builtin_amdgcn_wmma_f32_16x16x4_f32(
      /*neg_a=*/false, a, /*neg_b=*/false, b,
      /*c_mod=*/(short)0, c, /*reuse_a=*/false, /*reuse_b=*/false);
}

// Branch-free tanh: 1 - 2/(e^{2x}+1). v_exp_f32 + v_rcp_f32, no EXEC
// divergence (unlike the OCML tanhf), saturates correctly to +/-1.
__device__ __forceinline__ float fast_tanh(float x) {
  float e = __expf(2.0f * x);
  return 1.0f - 2.0f * __builtin_amdgcn_rcpf(e + 1.0f);
}

// --------------------------------------------------------------------------
// Monotonic grid barrier (persistent-kernel step sync). Counter zeroed by
// k_init each launch; target grows by nblk per step so no reset race.
// --------------------------------------------------------------------------
__device__ __forceinline__ void grid_sync(unsigned* bar, unsigned nblk,
                                          unsigned stepIdx) {
  __threadfence();
  __syncthreads();
  if (threadIdx.x == 0) {
    unsigned target = (stepIdx + 1u) * nblk;
    atomicAdd(bar, 1u);
    while (*((volatile unsigned*)bar) < target) {
      __builtin_amdgcn_s_sleep(1);
    }
  }
  __syncthreads();
  __threadfence();
}

// --------------------------------------------------------------------------
// Init: zero the four h double-buffers and the barrier counters.
// --------------------------------------------------------------------------
__global__ void k_init(float* __restrict__ hbufs, unsigned* __restrict__ bars,
                       int nfloat) {
  int i = blockIdx.x * blockDim.x + threadIdx.x;
  if (i < nfloat) hbufs[i] = 0.0f;
  if (blockIdx.x == 0 && threadIdx.x < 128) bars[threadIdx.x] = 0u;
}

// --------------------------------------------------------------------------
// Projection GEMM: Out[M,N] = X[M,K] @ W[N,K]^T + bias1 (+ bias2).
// 2x2 register-blocked: one wave per 32x32 output tile (4 WMMA accumulators)
// -> 4 b64 loads feed 4 WMMAs (1 load per WMMA). K is compile-time so the
// K-loop unrolls. GATHER re-indexes X rows from [B,T,I] as time-major.
// --------------------------------------------------------------------------
template <int K, bool GATHER, bool BIAS2>
__global__ void k_proj(const float* __restrict__ X, const float* __restrict__ W,
                       const float* __restrict__ bias1,
                       const float* __restrict__ bias2,
                       float* __restrict__ Out, int nrowPairs, int ncolPairs) {
  const int lane = threadIdx.x & 31;
  const int wave = threadIdx.x >> 5;
  const int gw = blockIdx.x * 4 + wave;           // 4 waves / 128-thread block
  if (gw >= nrowPairs * ncolPairs) return;        // wave-uniform exit

  const int mp = gw / ncolPairs;
  const int np = gw % ncolPairs;
  const int m0 = mp * 32, n0 = np * 32;
  const int rm = lane & 15;                       // A-row / B-col / D-col idx
  const int hi8 = (lane >> 4) * 8;                // D rows r+8 for upper half
  const int khalf = (lane >> 4) * 2;              // K sub-slab per lane half

  const float *aRow0, *aRow1;
  if (GATHER) {
    int r0 = m0 + rm, r1 = m0 + 16 + rm;          // time-major rows
    aRow0 = X + ((size_t)(r0 & (B_ - 1)) * T_ + (r0 >> 6)) * I_;
    aRow1 = X + ((size_t)(r1 & (B_ - 1)) * T_ + (r1 >> 6)) * I_;
  } else {
    aRow0 = X + (size_t)(m0 + rm) * K;
    aRow1 = X + (size_t)(m0 + 16 + rm) * K;
  }
  const float* bRow0 = W + (size_t)(n0 + rm) * K;       // W row == col of W^T
  const float* bRow1 = W + (size_t)(n0 + 16 + rm) * K;

  float bs0 = bias1[n0 + rm] + (BIAS2 ? bias2[n0 + rm] : 0.0f);
  float bs1 = bias1[n0 + 16 + rm] + (BIAS2 ? bias2[n0 + 16 + rm] : 0.0f);
  v8f c00 = {bs0, bs0, bs0, bs0, bs0, bs0, bs0, bs0};
  v8f c01 = {bs1, bs1, bs1, bs1, bs1, bs1, bs1, bs1};
  v8f c10 = c00;
  v8f c11 = c01;

#pragma unroll 8
  for (int kb = 0; kb < K; kb += 4) {
    v2f a0 = *(const v2f*)(aRow0 + kb + khalf);
    v2f a1 = *(const v2f*)(aRow1 + kb + khalf);
    v2f b0 = *(const v2f*)(bRow0 + kb + khalf);
    v2f b1 = *(const v2f*)(bRow1 + kb + khalf);
    c00 = wmma_f32_k4(a0, b0, c00);
    c01 = wmma_f32_k4(a0, b1, c01);
    c10 = wmma_f32_k4(a1, b0, c10);
    c11 = wmma_f32_k4(a1, b1, c11);
  }

  const int ldOut = ncolPairs * 32;
#pragma unroll
  for (int r = 0; r < 8; ++r) {
    size_t row0 = (size_t)(m0 + r + hi8) * ldOut;
    size_t row1 = (size_t)(m0 + 16 + r + hi8) * ldOut;
    Out[row0 + n0 + rm]      = c00[r];
    Out[row0 + n0 + 16 + rm] = c01[r];
    Out[row1 + n0 + rm]      = c10[r];
    Out[row1 + n0 + 16 + rm] = c11[r];
  }
}

// --------------------------------------------------------------------------
// Persistent recurrence: h_{t+1} = tanh(pre[t] + h_t @ Whh^T).
// Grid = 32 blocks x 128 threads = 128 waves = one 16x16 tile of the
// [64 x 512] state per wave per step (NOT register-blocked: the step is
// latency-bound and parallelism is free, so we minimize per-wave critical
// path). The 4 waves of a block share a mtile and have CONSECUTIVE ntiles,
// so the block needs 64 consecutive rows of Whh: staged ONCE into LDS
// (128 KB) by the Tensor Data Mover, then all 512 steps read the B-operand
// from LDS. h is double-buffered in global with one grid barrier per step.
// --------------------------------------------------------------------------
extern __shared__ float smem[];  // 64 * 512 floats = 128 KB (dynamic LDS)

__global__ void k_rnn(const float* __restrict__ pre,
                      const float* __restrict__ Whh,
                      float* h0, float* h1, float* __restrict__ ysOut,
                      unsigned* __restrict__ bar) {
  const int lane = threadIdx.x & 31;
  const int wave = threadIdx.x >> 5;
  const int gw = blockIdx.x * 4 + wave;   // 0..127
  const int mtile = gw >> 5;              // 0..3   (batch tiles, M=64)
  const int ntile = gw & 31;              // 0..31  (hidden tiles, N=512)
  const int m0 = mtile * 16, n0 = ntile * 16;
  const int rm = lane & 15;
  const int hi8 = (lane >> 4) * 8;
  const int khalf = (lane >> 4) * 2;

  // ---- TDM: stage this block's 64x512 f32 slab of Whh into LDS ----------
  const int rowBase = ((blockIdx.x * 4) & 31) * 16;
  if (wave == 0) {
    const unsigned long long ga =
        (unsigned long long)(uintptr_t)(Whh + (size_t)rowBase * H_);
    const unsigned int ldsOff = (unsigned int)(uintptr_t)&smem[0];

    u32x4 g0;
    g0.x = 1u;                                    // count=1, user descriptor
    g0.y = ldsOff;                                // D#.lds_addr
    g0.z = (unsigned int)ga;                      // global_addr[31:0]
    g0.w = (unsigned int)((ga >> 32) & 0x01FFFFFFu) | (2u << 30);  // type=2

    i32x8 g1;
    g1[0] = (int)(2u << 16);                      // data_size = 4 bytes
    g1[1] = (int)((512u & 0xFFFFu) << 16);        // tensor_dim0[15:0]=512
    g1[2] = (int)((512u >> 16) | (64u << 16));    // dim0[31:16] | dim1[15:0]=64
    g1[3] = (int)((64u >> 16) | (512u << 16));    // dim1[31:16] | tile_dim0=512
    g1[4] = (int)64;                              // tile_dim1=64, tile_dim2=0
    g1[5] = (int)512;                             // tensor_dim0_stride = 512
    g1[6] = 0;
    g1[7] = 0;

    i32x4 g2 = {0, 0, 0, 0};                      // 2-D tensor: groups 2/3 null
    i32x4 g3 = {0, 0, 0, 0};
    i32x8 g4 = {0, 0, 0, 0, 0, 0, 0, 0};          // clang-23 6-arg form
    __builtin_amdgcn_tensor_load_to_lds(g0, g1, g2, g3, g4, 0);
    __builtin_amdgcn_s_wait_tensorcnt(0);
  }
  __syncthreads();                                // release other 3 waves

  // This wave's B-operand row inside the LDS slab.
  const int wb = (wave * 16 + rm) * H_;

  float* cur = h0;
  float* nxt = h1;

  for (int t = 0; t < T_; ++t) {
    const float* preT = pre + (size_t)t * B_ * H_;
    v8f c = {};
#pragma unroll
    for (int r = 0; r < 8; ++r)
      c[r] = preT[(size_t)(m0 + r + hi8) * H_ + n0 + rm];

    const float* aRow = cur + (size_t)(m0 + rm) * H_;
#pragma unroll 16
    for (int kb = 0; kb < H_; kb += 4) {
      v2f a = *(const v2f*)(aRow + kb + khalf);
      v2f b = *(const v2f*)&smem[wb + kb + khalf];   // LDS B-operand
      c = wmma_f32_k4(a, b, c);
    }

#pragma unroll
    for (int r = 0; r < 8; ++r) {
      float v = fast_tanh(c[r]);
      nxt[(size_t)(m0 + r + hi8) * H_ + n0 + rm] = v;
      if (ysOut)
        ysOut[((size_t)t * B_ + (m0 + r + hi8)) * H_ + n0 + rm] = v;
    }

    grid_sync(bar, gridDim.x, (unsigned)t);
    float* tmp = cur; cur = nxt; nxt = tmp;
  }
  // T_ = 512 (even) -> final state ends up back in h0.
}

// --------------------------------------------------------------------------
// Host: phase sequence on one stream (graph-capture safe: no mallocs/syncs).
// Workspace layout:
//   [0, 64MB)      pre   (pre0, later overwritten by pre1)   T*B*H f32
//   [64MB,128MB)   ys0                                        T*B*H f32
//   [128MB, +512K) h0A,h0B,h1A,h1B                            4*B*H f32
//   then           barrier counters (bar[0] layer0, bar[64] layer1)
// --------------------------------------------------------------------------
extern "C" void kernel_launch(void* const* d_in, const int* in_sizes, int n_in,
                              void* d_out, int out_size, void* d_ws,
                              size_t ws_size, hipStream_t stream) {
  (void)in_sizes; (void)n_in; (void)out_size; (void)ws_size;

  const float* x     = (const float*)d_in[0];
  const float* W_ih0 = (const float*)d_in[1];
  const float* W_hh0 = (const float*)d_in[2];
  const float* b_ih0 = (const float*)d_in[3];
  const float* b_hh0 = (const float*)d_in[4];
  const float* W_ih1 = (const float*)d_in[5];
  const float* W_hh1 = (const float*)d_in[6];
  const float* b_ih1 = (const float*)d_in[7];
  const float* b_hh1 = (const float*)d_in[8];
  const float* W_fc  = (const float*)d_in[9];
  const float* b_fc  = (const float*)d_in[10];
  float* out = (float*)d_out;

  char* ws = (char*)d_ws;
  const size_t TBH = (size_t)T_ * B_ * H_;
  float* pre = (float*)ws;
  float* ys0 = (float*)(ws + TBH * sizeof(float));
  float* hb  = (float*)(ws + 2 * TBH * sizeof(float));
  unsigned* bars =
      (unsigned*)(ws + 2 * TBH * sizeof(float) + 4 * (size_t)B_ * H_ * sizeof(float));

  float* h0A = hb;
  float* h0B = hb + (size_t)B_ * H_;
  float* h1A = hb + 2 * (size_t)B_ * H_;
  float* h1B = hb + 3 * (size_t)B_ * H_;

  const size_t ldsBytes = 64 * 512 * sizeof(float);  // 128 KB TDM slab

  // 0) zero h buffers + barrier counters (deterministic graph replays)
  k_init<<<512, 256, 0, stream>>>(hb, bars, 4 * B_ * H_);

  // 1) pre0 = time-major(x) @ W_ih0^T + b_ih0 + b_hh0   [32768 x 512], K=256
  //    32x32 tiles: 1024 row-pairs x 16 col-pairs = 16384 waves
  k_proj<I_, true, true><<<(1024 * 16) / 4, 128, 0, stream>>>(
      x, W_ih0, b_ih0, b_hh0, pre, 1024, 16);

  // 2) layer-0 recurrence (512 steps), writes ys0
  k_rnn<<<32, 128, ldsBytes, stream>>>(pre, W_hh0, h0A, h0B, ys0, bars + 0);

  // 3) pre1 = ys0 @ W_ih1^T + b_ih1 + b_hh1             [32768 x 512], K=512
  k_proj<H_, false, true><<<(1024 * 16) / 4, 128, 0, stream>>>(
      ys0, W_ih1, b_ih1, b_hh1, pre, 1024, 16);

  // 4) layer-1 recurrence (512 steps), final h lands in h1A
  k_rnn<<<32, 128, ldsBytes, stream>>>(pre, W_hh1, h1A, h1B, nullptr, bars + 64);

  // 5) out = h_last @ W_fc^T + b_fc                     [64 x 256], K=512
  //    2 row-pairs x 8 col-pairs = 16 waves = 4 blocks
  k_proj<H_, false, false><<<4, 128, 0, stream>>>(
      h1A, W_fc, b_fc, nullptr, out, 2, 8);
}